// BlockSparseTensor_17497696763962
// MI455X (gfx1250) — compile-verified
//
#include <hip/hip_runtime.h>

// ---------------------------------------------------------------------------
// BlockSparseTensor.to_dense() for MI455X (gfx1250).
//
// Blocks are B-aligned (indices in [0, H/B)), so scatter-add == per-tile
// gather (deterministic, no global atomics). Memory-bound: read 256 MB of
// blocks + write 512 MB dense -> ~33 us at 23.3 TB/s. Phase 2 streams block
// data through LDS with the CDNA5 async copy engine
// (global_load_async_to_lds_b128 / ASYNCcnt), 4-deep pipelined (3 chunks in
// flight per wave = 1.5 KB/wave, ~48 KB in flight per WGP at 32 waves),
// accumulates in VGPRs, and writes the dense output with non-temporal stores
// (512 MB write-once stream; keep it out of the 192 MB L2).
// ---------------------------------------------------------------------------

typedef float  f32x4 __attribute__((ext_vector_type(4)));
typedef int    i32x4 __attribute__((ext_vector_type(4)));

#define BSZ        64
#define KS         8
#define GRID_T     64                  // H / BSZ
#define NTILES     (GRID_T * GRID_T)   // 4096
#define TILE_F     (BSZ * BSZ * KS)    // 32768 floats = 128 KB
#define THREADS    1024
#define CHUNK_F    (THREADS * 4)       // 4096 floats = 16 KB (1 float4/thread)
#define CHUNKS     (TILE_F / CHUNK_F)  // 8
#define NBUF       4                   // async pipeline depth (64 KB LDS)
#define MAXN_LDS   4096

// ---------------------------------------------------------------------------
// Phase 1: deterministic per-tile block lists.
//   offsets[t]   = exclusive prefix of per-tile counts (4097 ints)
//   list[o..o+c) = block ids hitting tile t, in ascending block order
// ---------------------------------------------------------------------------
__global__ __launch_bounds__(THREADS) void build_tile_lists(
    const int* __restrict__ bi, int N,
    int* __restrict__ offsets, int* __restrict__ list) {
  __shared__ int tileid[MAXN_LDS];
  __shared__ int counts[NTILES];   // becomes per-tile exclusive offsets later
  __shared__ int tsum[THREADS];
  const int t = threadIdx.x;

  for (int i = t; i < NTILES; i += THREADS) counts[i] = 0;
  for (int n = t; n < N; n += THREADS) {
    const int gr = bi[2 * n + 0];
    const int gc = bi[2 * n + 1];
    tileid[n] = (gr >= 0 && gr < GRID_T && gc >= 0 && gc < GRID_T)
                    ? gr * GRID_T + gc : -1;   // fully-outside blocks dropped
  }
  __syncthreads();

  for (int n = t; n < N; n += THREADS) {
    const int tid_ = tileid[n];
    if (tid_ >= 0) atomicAdd(&counts[tid_], 1);  // counts order-independent
  }
  __syncthreads();

  // 2-level exclusive scan over 4096 counts (4 per thread + 1024-wide scan)
  const int c0 = counts[4 * t + 0], c1 = counts[4 * t + 1];
  const int c2 = counts[4 * t + 2], c3 = counts[4 * t + 3];
  tsum[t] = c0 + c1 + c2 + c3;
  __syncthreads();
  for (int d = 1; d < THREADS; d <<= 1) {
    const int v = (t >= d) ? tsum[t - d] : 0;
    __syncthreads();
    tsum[t] += v;
    __syncthreads();
  }
  int run = (t > 0) ? tsum[t - 1] : 0;
  counts[4 * t + 0] = run; offsets[4 * t + 0] = run; run += c0;
  counts[4 * t + 1] = run; offsets[4 * t + 1] = run; run += c1;
  counts[4 * t + 2] = run; offsets[4 * t + 2] = run; run += c2;
  counts[4 * t + 3] = run; offsets[4 * t + 3] = run;
  if (t == THREADS - 1) offsets[NTILES] = tsum[THREADS - 1];
  __syncthreads();

  // Rank = #earlier blocks with the same tile -> deterministic list order.
  for (int n = t; n < N; n += THREADS) {
    const int tile = tileid[n];
    if (tile < 0) continue;
    int rank = 0, m = 0;
    for (; m + 4 <= n; m += 4) {
      const i32x4 q = *(const i32x4*)&tileid[m];
      rank += (q.x == tile) + (q.y == tile) + (q.z == tile) + (q.w == tile);
    }
    for (; m < n; ++m) rank += (tileid[m] == tile);
    list[counts[tile] + rank] = n;
  }
}

// ---------------------------------------------------------------------------
// Phase 2: one workgroup per tile. Thread owns 8 float4s of the tile
// (f = c*4096 + tid*4). Blocks stream global -> LDS via async copies,
// 4-deep pipelined through 64 KB LDS; accumulate in VGPRs; NT-store the
// tile. Per wave: each thread reads only the LDS slot its own async copy
// wrote, so s_wait_asynccnt alone orders producer->consumer (no barriers).
// In-order completion => asynccnt<=3 retires the oldest of 4 in flight.
// ---------------------------------------------------------------------------
__global__ __launch_bounds__(THREADS) void accumulate_tiles(
    const float* __restrict__ bv,
    const int* __restrict__ offsets,
    const int* __restrict__ list,
    float* __restrict__ out) {
  __shared__ f32x4 stage[NBUF][THREADS];   // 64 KB, 4-deep pipeline

  const int tid  = threadIdx.x;
  const int tile = blockIdx.x;
  const int gr   = tile >> 6;
  const int gc   = tile & 63;
  const int off  = offsets[tile];
  const int cnt  = offsets[tile + 1] - off;

  f32x4 acc[CHUNKS];
#pragma unroll
  for (int i = 0; i < CHUNKS; ++i) acc[i] = (f32x4)0.0f;

  // Issue one 16-byte/lane async copy: block n, chunk c -> stage[b][tid].
  auto issue = [&](int n, int c, int b) {
    const float* g = bv + (size_t)n * TILE_F + c * CHUNK_F + tid * 4;
    const unsigned lds = (unsigned)(size_t)(&stage[b][tid]);
    asm volatile("global_load_async_to_lds_b128 %0, %1, off"
                 :: "v"(lds), "v"(g) : "memory");
  };

  if (cnt > 0) {
    int nCur = list[off];
    // Prologue: chunks 0..2 of the first block (3 copies in flight).
    issue(nCur, 0, 0);
    issue(nCur, 1, 1);
    issue(nCur, 2, 2);

    // Main: all blocks except the last. it = j*8+c; buffer = c&3 (8%4==0).
    for (int j = 0; j < cnt - 1; ++j) {
      const int nNext = list[off + j + 1];
#pragma unroll
      for (int c = 0; c < CHUNKS; ++c) {
        if (c < 5) issue(nCur, c + 3, (c + 3) & 3);        // same block
        else       issue(nNext, c - 5, (c + 3) & 3);       // next block
        asm volatile("s_wait_asynccnt 0x3" ::: "memory");  // oldest retired
        const f32x4 v = stage[c & 3][tid];                 // ds_load_b128
        acc[c] += v;
      }
      nCur = nNext;
    }

    // Tail: last block; drain with compile-time wait levels.
#pragma unroll
    for (int c = 0; c < CHUNKS; ++c) {
      if (c < 5) {
        issue(nCur, c + 3, (c + 3) & 3);
        asm volatile("s_wait_asynccnt 0x3" ::: "memory");
      } else if (c == 5) {
        asm volatile("s_wait_asynccnt 0x2" ::: "memory");
      } else if (c == 6) {
        asm volatile("s_wait_asynccnt 0x1" ::: "memory");
      } else {
        asm volatile("s_wait_asynccnt 0x0" ::: "memory");
      }
      const f32x4 v = stage[c & 3][tid];
      acc[c] += v;
    }
  }

  // Store full tile (zero tiles included -> d_out fully initialized).
#pragma unroll
  for (int i = 0; i < CHUNKS; ++i) {
    const int f   = i * CHUNK_F + tid * 4;   // flat float index inside tile
    const int row = f >> 9;                  // 512 floats per tile row
    const int w   = f & 511;
    float* dst = out + (((size_t)(gr * BSZ + row) * (GRID_T * BSZ)) +
                        gc * BSZ) * KS + w;
    __builtin_nontemporal_store(acc[i], (f32x4*)dst);
  }
}

// ---------------------------------------------------------------------------
extern "C" void kernel_launch(void* const* d_in, const int* in_sizes, int n_in,
                              void* d_out, int out_size, void* d_ws, size_t ws_size,
                              hipStream_t stream) {
  const float* bv  = (const float*)d_in[0];  // [N, 64, 64, 8] f32
  const int*   bi  = (const int*)d_in[1];    // [N, 2] int
  float*       out = (float*)d_out;          // [4096, 4096, 8] f32

  int N = in_sizes[1] / 2;
  if (N > MAXN_LDS) N = MAXN_LDS;            // phase-1 LDS capacity guard

  int* offsets = (int*)d_ws;                 // 4097 ints
  int* list    = offsets + (NTILES + 1);     // N ints

  build_tile_lists<<<dim3(1), dim3(THREADS), 0, stream>>>(bi, N, offsets, list);
  accumulate_tiles<<<dim3(NTILES), dim3(THREADS), 0, stream>>>(bv, offsets, list, out);
}